// HessianAttentionLayer_85177791414418
// MI455X (gfx1250) — compile-verified
//
#include <hip/hip_runtime.h>
#include <math.h>

#define Nn 4096
#define Dd 256
#define Ss 32
#define Kk 4
#define ETA 0.5f
#define EPS 0.5f

typedef __attribute__((ext_vector_type(2))) float v2f;
typedef __attribute__((ext_vector_type(8))) float v8f;

// ---------------------------------------------------------------------------
// 0. zero the two scalar-loss slots of d_out
// ---------------------------------------------------------------------------
__global__ void init_scalars_kernel(float* out) {
    if (threadIdx.x == 0) {
        out[(size_t)Nn * Dd + 0] = 0.0f;
        out[(size_t)Nn * Dd + 1] = 0.0f;
    }
}

// ---------------------------------------------------------------------------
// 1. Z[k][s][n] = sum_d U[k][d][s] * H[n][d]
// grid: K*S*(N/256) blocks x 256 threads
// ---------------------------------------------------------------------------
__global__ void compute_Z_kernel(const float* __restrict__ U,
                                 const float* __restrict__ H,
                                 float* __restrict__ Z) {
    int b = blockIdx.x;
    int nblk = b & 15;            // N/256 = 16
    int rem  = b >> 4;
    int s = rem & (Ss - 1);
    int k = rem / Ss;
    int n = nblk * 256 + threadIdx.x;
    const float* Uk   = U + (size_t)k * Dd * Ss;
    const float* Hrow = H + (size_t)n * Dd;
    float acc = 0.0f;
    #pragma unroll 4
    for (int d = 0; d < Dd; ++d)
        acc += Uk[d * Ss + s] * Hrow[d];
    Z[((size_t)k * Ss + s) * Nn + n] = acc;
}

// ---------------------------------------------------------------------------
// 2. M[k] = I + coeff * Z[k] Z[k]^T   (32x32 per k)
// grid: K*S blocks x 32 threads (thread = column t)
// ---------------------------------------------------------------------------
__global__ __launch_bounds__(32) void compute_M_kernel(const float* __restrict__ Z,
                                                       float* __restrict__ M) {
    int k = blockIdx.x / Ss;
    int s = blockIdx.x % Ss;
    int t = threadIdx.x;
    const float* Zs = Z + ((size_t)k * Ss + s) * Nn;
    const float* Zt = Z + ((size_t)k * Ss + t) * Nn;
    float acc = 0.0f;
    for (int n = 0; n < Nn; ++n) acc += Zs[n] * Zt[n];
    const float coeff = (float)Ss / ((float)Nn * EPS * EPS);
    M[((size_t)k * Ss + s) * Ss + t] = (s == t ? 1.0f : 0.0f) + coeff * acc;
}

// ---------------------------------------------------------------------------
// 3. Minv[k] = M[k]^-1 via Cholesky (serial in one lane, trivial work)
// grid: K blocks x 32 threads
// ---------------------------------------------------------------------------
__global__ __launch_bounds__(32) void chol_inv_kernel(const float* __restrict__ M,
                                                      float* __restrict__ Minv) {
    __shared__ float A[Ss * Ss];
    __shared__ float Xl[Ss * Ss];
    int k = blockIdx.x;
    for (int i = threadIdx.x; i < Ss * Ss; i += 32) {
        A[i] = M[(size_t)k * Ss * Ss + i];
        Xl[i] = 0.0f;
    }
    __syncthreads();
    if (threadIdx.x == 0) {
        // in-place lower Cholesky
        for (int j = 0; j < Ss; ++j) {
            float s = A[j * Ss + j];
            for (int p = 0; p < j; ++p) s -= A[j * Ss + p] * A[j * Ss + p];
            float dj = sqrtf(s);
            A[j * Ss + j] = dj;
            for (int i = j + 1; i < Ss; ++i) {
                float v = A[i * Ss + j];
                for (int p = 0; p < j; ++p) v -= A[i * Ss + p] * A[j * Ss + p];
                A[i * Ss + j] = v / dj;
            }
        }
        // Xl = L^-1 (lower triangular)
        for (int j = 0; j < Ss; ++j) {
            Xl[j * Ss + j] = 1.0f / A[j * Ss + j];
            for (int i = j + 1; i < Ss; ++i) {
                float v = 0.0f;
                for (int p = j; p < i; ++p) v += A[i * Ss + p] * Xl[p * Ss + j];
                Xl[i * Ss + j] = -v / A[i * Ss + i];
            }
        }
    }
    __syncthreads();
    // Minv = Xl^T Xl
    for (int e = threadIdx.x; e < Ss * Ss; e += 32) {
        int a = e / Ss, b = e % Ss;
        int p0 = a > b ? a : b;
        float v = 0.0f;
        for (int p = p0; p < Ss; ++p) v += Xl[p * Ss + a] * Xl[p * Ss + b];
        Minv[(size_t)k * Ss * Ss + e] = v;
    }
}

// ---------------------------------------------------------------------------
// 4. W[k][s][n] = sum_t Minv[k][s][t] * Z[k][t][n]
// grid: K*S*(N/256) blocks x 256 threads
// ---------------------------------------------------------------------------
__global__ void compute_W_kernel(const float* __restrict__ Minv,
                                 const float* __restrict__ Z,
                                 float* __restrict__ W) {
    int b = blockIdx.x;
    int nblk = b & 15;
    int rem  = b >> 4;
    int s = rem & (Ss - 1);
    int k = rem / Ss;
    int n = nblk * 256 + threadIdx.x;
    const float* Mi = Minv + ((size_t)k * Ss + s) * Ss;
    float acc = 0.0f;
    #pragma unroll
    for (int t = 0; t < Ss; ++t)
        acc += Mi[t] * Z[((size_t)k * Ss + t) * Nn + n];
    W[((size_t)k * Ss + s) * Nn + n] = acc;
}

// ---------------------------------------------------------------------------
// 5. Fused flash attention: scores = Z^T W (per k), mask, online softmax,
//    Zalpha[k][s][i] = sum_j alpha[i][j] * Z[k][s][j]
// One wave (32 threads) per 16-row i-tile per k.  grid: K * N/16 = 1024
// All matrix math on V_WMMA_F32_16X16X4_F32; next-tile streams prefetched
// with global_prefetch_b8.
// ---------------------------------------------------------------------------
__global__ __launch_bounds__(32) void flash_attn_kernel(const float* __restrict__ Z,
                                                        const float* __restrict__ W,
                                                        const float* __restrict__ mask,
                                                        float* __restrict__ Zalpha) {
    const int wid  = blockIdx.x;
    const int k    = wid >> 8;              // N/16 = 256 tiles per k
    const int i0   = (wid & 255) * 16;
    const int lane = threadIdx.x;
    const int half = lane >> 4;
    const int lh   = lane & 15;

    const float* Zk = Z + (size_t)k * Ss * Nn;
    const float* Wk = W + (size_t)k * Ss * Nn;

    __shared__ float P_lds[16 * 17];        // padded: conflict-free transpose
    __shared__ float bcast[16];

    // Loop-invariant A fragments: A[i][s] = Z[s][i0+i]  (16 x 32, K-chunks of 4)
    v2f aZ[8];
    #pragma unroll
    for (int sc = 0; sc < 8; ++sc) {
        int s = sc * 4 + 2 * half;
        aZ[sc].x = Zk[(size_t)(s + 0) * Nn + i0 + lh];
        aZ[sc].y = Zk[(size_t)(s + 1) * Nn + i0 + lh];
    }

    v8f outT0 = {};                          // Out^T rows s=0..15
    v8f outT1 = {};                          // Out^T rows s=16..31
    float m[8], l[8];
    #pragma unroll
    for (int r = 0; r < 8; ++r) { m[r] = -1.0e30f; l[r] = 0.0f; }

    for (int j0 = 0; j0 < Nn; j0 += 16) {
        // ---- prefetch next tile's streams (global_prefetch_b8)
        if (j0 + 16 < Nn) {
            __builtin_prefetch(&Wk[(size_t)lane * Nn + j0 + 16], 0, 3);
            __builtin_prefetch(&Zk[(size_t)lane * Nn + j0 + 16], 0, 3);
            __builtin_prefetch(&mask[(size_t)(i0 + lh) * Nn + j0 + 16], 0, 3);
        }
        // ---- scores tile (16 x 16), C-layout: c[r] <-> row i0+r+8*half, col j0+lh
        v8f c = {};
        #pragma unroll
        for (int sc = 0; sc < 8; ++sc) {
            int s = sc * 4 + 2 * half;
            v2f b;
            b.x = Wk[(size_t)(s + 0) * Nn + j0 + lh];
            b.y = Wk[(size_t)(s + 1) * Nn + j0 + lh];
            c = __builtin_amdgcn_wmma_f32_16x16x4_f32(false, aZ[sc], false, b,
                                                      (short)0, c, false, false);
        }
        // ---- mask penalty
        #pragma unroll
        for (int r = 0; r < 8; ++r) {
            int i = i0 + r + 8 * half;
            float mv = mask[(size_t)i * Nn + j0 + lh];
            c[r] += (1.0f - mv) * -1.0e9f;
        }
        // ---- row max across the 16 lanes of each half
        float t[8], p[8], rs[8], scale[8];
        #pragma unroll
        for (int r = 0; r < 8; ++r) t[r] = c[r];
        #pragma unroll
        for (int off = 1; off < 16; off <<= 1)
            #pragma unroll
            for (int r = 0; r < 8; ++r)
                t[r] = fmaxf(t[r], __shfl_xor(t[r], off, 32));
        #pragma unroll
        for (int r = 0; r < 8; ++r) {
            float mn = fmaxf(m[r], t[r]);
            scale[r] = __expf(m[r] - mn);
            m[r] = mn;
            p[r] = __expf(c[r] - mn);
            rs[r] = p[r];
        }
        #pragma unroll
        for (int off = 1; off < 16; off <<= 1)
            #pragma unroll
            for (int r = 0; r < 8; ++r)
                rs[r] += __shfl_xor(rs[r], off, 32);
        #pragma unroll
        for (int r = 0; r < 8; ++r) l[r] = l[r] * scale[r] + rs[r];

        // ---- broadcast per-row scale and stage P (transposed read later)
        __syncthreads();
        if (lane == 0)
            #pragma unroll
            for (int r = 0; r < 8; ++r) bcast[r] = scale[r];
        if (lane == 16)
            #pragma unroll
            for (int r = 0; r < 8; ++r) bcast[8 + r] = scale[r];
        #pragma unroll
        for (int r = 0; r < 8; ++r)
            P_lds[(r + 8 * half) * 17 + lh] = p[r];
        __syncthreads();

        float s_i = bcast[lh];               // scale for column i = i0+lh of Out^T
        #pragma unroll
        for (int r = 0; r < 8; ++r) { outT0[r] *= s_i; outT1[r] *= s_i; }

        // ---- Out^T += Z_tile * P^T  (K dim = j, chunks of 4)
        #pragma unroll
        for (int jc = 0; jc < 16; jc += 4) {
            v2f bp, a0, a1;
            bp.x = P_lds[lh * 17 + jc + 2 * half + 0];
            bp.y = P_lds[lh * 17 + jc + 2 * half + 1];
            a0.x = Zk[(size_t)(lh) * Nn + j0 + jc + 2 * half + 0];
            a0.y = Zk[(size_t)(lh) * Nn + j0 + jc + 2 * half + 1];
            a1.x = Zk[(size_t)(16 + lh) * Nn + j0 + jc + 2 * half + 0];
            a1.y = Zk[(size_t)(16 + lh) * Nn + j0 + jc + 2 * half + 1];
            outT0 = __builtin_amdgcn_wmma_f32_16x16x4_f32(false, a0, false, bp,
                                                          (short)0, outT0, false, false);
            outT1 = __builtin_amdgcn_wmma_f32_16x16x4_f32(false, a1, false, bp,
                                                          (short)0, outT1, false, false);
        }
    }

    // ---- normalize by softmax denominator and store Zalpha
    __syncthreads();
    if (lane == 0)
        #pragma unroll
        for (int r = 0; r < 8; ++r) bcast[r] = 1.0f / l[r];
    if (lane == 16)
        #pragma unroll
        for (int r = 0; r < 8; ++r) bcast[8 + r] = 1.0f / l[r];
    __syncthreads();
    float linv = bcast[lh];
    float* Zak = Zalpha + (size_t)k * Ss * Nn;
    #pragma unroll
    for (int r = 0; r < 8; ++r) {
        Zak[(size_t)(r + 8 * half) * Nn + i0 + lh]       = outT0[r] * linv;
        Zak[(size_t)(16 + r + 8 * half) * Nn + i0 + lh]  = outT1[r] * linv;
    }
}

// ---------------------------------------------------------------------------
// 6. UMt[k][t][d] = sum_s U[k][d][s] * Minv[k][s][t]
// grid: K*S blocks x 256 threads (d)
// ---------------------------------------------------------------------------
__global__ void compute_UMt_kernel(const float* __restrict__ U,
                                   const float* __restrict__ Minv,
                                   float* __restrict__ UMt) {
    int k = blockIdx.x / Ss;
    int t = blockIdx.x % Ss;
    int d = threadIdx.x;
    float acc = 0.0f;
    #pragma unroll
    for (int s = 0; s < Ss; ++s)
        acc += U[((size_t)k * Dd + d) * Ss + s] * Minv[((size_t)k * Ss + s) * Ss + t];
    UMt[((size_t)k * Ss + t) * Dd + d] = acc;
}

// ---------------------------------------------------------------------------
// 7. Y = L @ X   (N x N) @ (N x D), fp32 WMMA.
// One wave computes a 16x64 strip (4 accumulator fragments) so each strided
// A-fragment of L is reused by 4 WMMAs.  grid: (N/16)*(D/64) = 1024 x 32.
// ---------------------------------------------------------------------------
__global__ __launch_bounds__(32) void gemm_LX_kernel(const float* __restrict__ L,
                                                     const float* __restrict__ X,
                                                     float* __restrict__ Y) {
    int tile = blockIdx.x;
    int i0 = (tile >> 2) * 16;               // D/64 = 4 strips
    int d0 = (tile & 3) * 64;
    int lane = threadIdx.x;
    int half = lane >> 4;
    int lh   = lane & 15;
    v8f acc0 = {}, acc1 = {}, acc2 = {}, acc3 = {};
    const float* Lrow = L + (size_t)(i0 + lh) * Nn + 2 * half;
    for (int kk = 0; kk < Nn; kk += 4) {
        v2f a;
        a.x = Lrow[kk + 0];
        a.y = Lrow[kk + 1];
        const float* Xr0 = X + (size_t)(kk + 2 * half) * Dd + d0 + lh;
        const float* Xr1 = Xr0 + Dd;
        v2f b0, b1, b2, b3;
        b0.x = Xr0[0];   b0.y = Xr1[0];
        b1.x = Xr0[16];  b1.y = Xr1[16];
        b2.x = Xr0[32];  b2.y = Xr1[32];
        b3.x = Xr0[48];  b3.y = Xr1[48];
        acc0 = __builtin_amdgcn_wmma_f32_16x16x4_f32(false, a, false, b0,
                                                     (short)0, acc0, false, false);
        acc1 = __builtin_amdgcn_wmma_f32_16x16x4_f32(false, a, false, b1,
                                                     (short)0, acc1, false, false);
        acc2 = __builtin_amdgcn_wmma_f32_16x16x4_f32(false, a, false, b2,
                                                     (short)0, acc2, false, false);
        acc3 = __builtin_amdgcn_wmma_f32_16x16x4_f32(false, a, false, b3,
                                                     (short)0, acc3, false, false);
    }
    #pragma unroll
    for (int r = 0; r < 8; ++r) {
        float* Yr = Y + (size_t)(i0 + r + 8 * half) * Dd + d0 + lh;
        Yr[0]  = acc0[r];
        Yr[16] = acc1[r];
        Yr[32] = acc2[r];
        Yr[48] = acc3[r];
    }
}

// ---------------------------------------------------------------------------
// 8. H_out[n][d] = softthresh( H + ETA*agg - ETA*lambda*LH )
//    agg[n][d] = sum_k sum_t UMt[k][t][d] * Zalpha[k][t][n]
// grid: N blocks x 256 threads (d)
// ---------------------------------------------------------------------------
__global__ void combine_kernel(const float* __restrict__ H,
                               const float* __restrict__ UMt,
                               const float* __restrict__ Zalpha,
                               const float* __restrict__ LH,
                               const float* __restrict__ lambda_lap,
                               const float* __restrict__ threshold,
                               float* __restrict__ Hout) {
    int n = blockIdx.x;
    int d = threadIdx.x;
    float agg = 0.0f;
    #pragma unroll
    for (int k = 0; k < Kk; ++k)
        #pragma unroll
        for (int t = 0; t < Ss; ++t)
            agg += UMt[((size_t)k * Ss + t) * Dd + d] *
                   Zalpha[((size_t)k * Ss + t) * Nn + n];
    float lam = lambda_lap[0];
    float hh  = H[(size_t)n * Dd + d] + ETA * agg - ETA * lam * LH[(size_t)n * Dd + d];
    float av  = fabsf(hh) - threshold[d];
    Hout[(size_t)n * Dd + d] = (av > 0.0f) ? copysignf(av, hh) : 0.0f;
}

// ---------------------------------------------------------------------------
// 9. lap_smooth = sum(H_out * (L @ H_out))   -> out[N*D + 1]
// ---------------------------------------------------------------------------
__global__ void lap_smooth_kernel(const float* __restrict__ Hout,
                                  const float* __restrict__ LH2,
                                  float* __restrict__ out) {
    size_t i = (size_t)blockIdx.x * 256 + threadIdx.x;
    float v = Hout[i] * LH2[i];
    #pragma unroll
    for (int off = 1; off < 32; off <<= 1) v += __shfl_xor(v, off, 32);
    __shared__ float partial[8];
    if ((threadIdx.x & 31) == 0) partial[threadIdx.x >> 5] = v;
    __syncthreads();
    if (threadIdx.x == 0) {
        float s = 0.0f;
        #pragma unroll
        for (int w = 0; w < 8; ++w) s += partial[w];
        atomicAdd(&out[(size_t)Nn * Dd + 1], s);
    }
}

// ---------------------------------------------------------------------------
// 10. orth_loss -> out[N*D]
// grid: K*K*S*S/256 = 64 blocks x 256 threads
// ---------------------------------------------------------------------------
__global__ void orth_loss_kernel(const float* __restrict__ U,
                                 float* __restrict__ out) {
    int idx = blockIdx.x * 256 + threadIdx.x;
    int e = idx % Ss;
    int rem = idx / Ss;
    int s = rem % Ss; rem /= Ss;
    int l = rem % Kk;
    int k = rem / Kk;
    float g = 0.0f;
    #pragma unroll 4
    for (int d = 0; d < Dd; ++d)
        g += U[((size_t)k * Dd + d) * Ss + s] * U[((size_t)l * Dd + d) * Ss + e];
    float contrib = 0.0f;
    if (k == l) { float dl = g - (s == e ? 1.0f : 0.0f); contrib = dl * dl; }
    else if (k < l) contrib = g * g;
    float v = contrib;
    #pragma unroll
    for (int off = 1; off < 32; off <<= 1) v += __shfl_xor(v, off, 32);
    __shared__ float partial[8];
    if ((threadIdx.x & 31) == 0) partial[threadIdx.x >> 5] = v;
    __syncthreads();
    if (threadIdx.x == 0) {
        float sum = 0.0f;
        #pragma unroll
        for (int w = 0; w < 8; ++w) sum += partial[w];
        atomicAdd(&out[(size_t)Nn * Dd], sum);
    }
}

// ---------------------------------------------------------------------------
extern "C" void kernel_launch(void* const* d_in, const int* in_sizes, int n_in,
                              void* d_out, int out_size, void* d_ws, size_t ws_size,
                              hipStream_t stream) {
    const float* H     = (const float*)d_in[0];
    const float* mask  = (const float*)d_in[1];
    const float* L     = (const float*)d_in[2];
    const float* U     = (const float*)d_in[3];
    const float* lam   = (const float*)d_in[4];
    const float* thr   = (const float*)d_in[5];
    float* out = (float*)d_out;

    float* ws = (float*)d_ws;
    const size_t KSN = (size_t)Kk * Ss * Nn;     // 524288
    float* Z      = ws;                          // KSN
    float* W      = Z + KSN;                     // KSN
    float* Zalpha = W + KSN;                     // KSN
    float* M      = Zalpha + KSN;                // K*S*S = 4096
    float* Minv   = M + (size_t)Kk * Ss * Ss;    // 4096
    float* UMt    = Minv + (size_t)Kk * Ss * Ss; // K*S*D = 32768
    float* LH     = UMt + (size_t)Kk * Ss * Dd;  // N*D
    float* LH2    = LH + (size_t)Nn * Dd;        // N*D

    init_scalars_kernel<<<1, 32, 0, stream>>>(out);
    compute_Z_kernel<<<Kk * Ss * (Nn / 256), 256, 0, stream>>>(U, H, Z);
    compute_M_kernel<<<Kk * Ss, 32, 0, stream>>>(Z, M);
    chol_inv_kernel<<<Kk, 32, 0, stream>>>(M, Minv);
    compute_W_kernel<<<Kk * Ss * (Nn / 256), 256, 0, stream>>>(Minv, Z, W);
    flash_attn_kernel<<<Kk * (Nn / 16), 32, 0, stream>>>(Z, W, mask, Zalpha);
    compute_UMt_kernel<<<Kk * Ss, Dd, 0, stream>>>(U, Minv, UMt);
    gemm_LX_kernel<<<(Nn / 16) * (Dd / 64), 32, 0, stream>>>(L, H, LH);
    combine_kernel<<<Nn, Dd, 0, stream>>>(H, UMt, Zalpha, LH, lam, thr, out);
    gemm_LX_kernel<<<(Nn / 16) * (Dd / 64), 32, 0, stream>>>(L, out, LH2);
    lap_smooth_kernel<<<(Nn * Dd) / 256, 256, 0, stream>>>(out, LH2, out);
    orth_loss_kernel<<<(Kk * Kk * Ss * Ss) / 256, 256, 0, stream>>>(U, out);
}